// PINNLeakDetectionGNN_58909771432751
// MI455X (gfx1250) — compile-verified
//
#include <hip/hip_runtime.h>

#define BATCH 8
#define RES_HEAD_C 100.0f

typedef float v2f __attribute__((ext_vector_type(2)));
typedef float v4f __attribute__((ext_vector_type(4)));
typedef float v8f __attribute__((ext_vector_type(8)));
typedef int   v2i __attribute__((ext_vector_type(2)));

// ---------------- kernel 0: zero workspace (net accumulators + loss scalar) ----
__global__ void pinn_zero_ws(float* __restrict__ ws, int n) {
    int i = (blockIdx.x * blockDim.x + threadIdx.x) * 4;
    if (i + 3 < n) {
        v4f z = {};
        *(v4f*)(ws + i) = z;
    } else {
        for (; i < n; ++i) ws[i] = 0.0f;
    }
}

// ---------------- kernel 1: per-edge flows + scatter-add of net flows ----------
// 2 edges per thread: b64 index loads, b128 attr load, b64 NT flow stores.
// flows_out[b*E+e] = cond[e] * (heads[b,src] - heads[b,dst])
// net[b*N+src] += flow ; net[b*N+dst] -= flow
__global__ void pinn_edge_kernel(const float* __restrict__ heads,
                                 const int*   __restrict__ edge_index,
                                 const float* __restrict__ edge_attr,
                                 float* __restrict__ flows_out,
                                 float* __restrict__ net,
                                 int E, int N) {
    long e0 = (long)(blockIdx.x * blockDim.x + threadIdx.x) * 2;
    if (e0 >= E) return;

    if (e0 + 1 < E) {
        // streamed once: non-temporal, keep L2 for heads/net
        v2i s2 = __builtin_nontemporal_load((const v2i*)(edge_index + e0));
        v2i d2 = __builtin_nontemporal_load((const v2i*)(edge_index + E + e0));
        v4f a4 = __builtin_nontemporal_load((const v4f*)(edge_attr + 2 * e0));
        float c0 = a4.x, c1 = a4.z;
#pragma unroll
        for (int b = 0; b < BATCH; ++b) {
            const float* hb = heads + (size_t)b * N;
            float*       nb = net   + (size_t)b * N;
            float f0 = c0 * (hb[s2.x] - hb[d2.x]);
            float f1 = c1 * (hb[s2.y] - hb[d2.y]);
            v2f fv; fv.x = f0; fv.y = f1;
            __builtin_nontemporal_store(fv, (v2f*)(flows_out + (size_t)b * E + e0));
            unsafeAtomicAdd(nb + s2.x,  f0);
            unsafeAtomicAdd(nb + d2.x, -f0);
            unsafeAtomicAdd(nb + s2.y,  f1);
            unsafeAtomicAdd(nb + d2.y, -f1);
        }
    } else {
        // odd tail edge (not hit for even E, kept for generality)
        int   src  = edge_index[e0];
        int   dst  = edge_index[E + e0];
        float cond = edge_attr[2 * e0];
#pragma unroll
        for (int b = 0; b < BATCH; ++b) {
            float flow = cond * (heads[(size_t)b * N + src] - heads[(size_t)b * N + dst]);
            __builtin_nontemporal_store(flow, flows_out + (size_t)b * E + e0);
            unsafeAtomicAdd(net + (size_t)b * N + src,  flow);
            unsafeAtomicAdd(net + (size_t)b * N + dst, -flow);
        }
    }
}

// ---------------- kernel 2: continuity loss sum via WMMA ones-matmul reduction -
// Per wave-iter: two b128 loads/lane (net, demands) -> 128 squared residuals ->
// two V_WMMA_F32_16X16X4_F32 (B = ones) accumulating row sums in C.
// sum(A) == sum over M of D[M,0] (N=0 column: lanes 0 and 16, VGPRs 0..7).
__global__ void pinn_continuity_wmma(const float* __restrict__ net,
                                     const float* __restrict__ demands,
                                     float* __restrict__ acc, int n) {
    int lane   = threadIdx.x & 31;
    int wave   = (blockIdx.x * blockDim.x + threadIdx.x) >> 5;
    int nWaves = (gridDim.x * blockDim.x) >> 5;

    long nMain = (long)n & ~127L;   // multiple of 128 handled by WMMA path

    v2f ones; ones.x = 1.0f; ones.y = 1.0f;
    v8f c0 = {};
    v8f c1 = {};

    // wave-uniform bounds, unguarded b128 loads: EXEC stays all-1s at the WMMAs
    for (long base = (long)wave * 128; base < nMain; base += (long)nWaves * 128) {
        long i = base + (long)lane * 4;
        v4f xn = *(const v4f*)(net + i);
        v4f xd = *(const v4f*)(demands + i);
        v4f r;
        r.x = xn.x - xd.x; r.y = xn.y - xd.y;
        r.z = xn.z - xd.z; r.w = xn.w - xd.w;
        v2f a0; a0.x = r.x * r.x; a0.y = r.y * r.y;
        v2f a1; a1.x = r.z * r.z; a1.y = r.w * r.w;
        // 8 args: (neg_a, A, neg_b, B, c_mod, C, reuse_a, reuse_b)
        c0 = __builtin_amdgcn_wmma_f32_16x16x4_f32(
                 false, a0, false, ones, (short)0, c0, false, false);
        c1 = __builtin_amdgcn_wmma_f32_16x16x4_f32(
                 false, a1, false, ones, (short)0, c1, false, false);
    }

    v8f cs = c0 + c1;
    float s = cs[0] + cs[1] + cs[2] + cs[3] + cs[4] + cs[5] + cs[6] + cs[7];
    float v0  = __shfl(s, 0, 32);    // M = 0..7  at N=0
    float v16 = __shfl(s, 16, 32);   // M = 8..15 at N=0
    if (lane == 0) unsafeAtomicAdd(acc, v0 + v16);

    // scalar tail (n % 128), wave 0 only
    if (wave == 0) {
        float tail = 0.0f;
        for (long i = nMain + lane; i < n; i += 32) {
            float x = net[i] - demands[i];
            tail += x * x;
        }
        unsafeAtomicAdd(acc, tail);
    }
}

// ---------------- kernel 3: boundary loss + finalize the three scalars --------
__global__ void pinn_finalize(const float* __restrict__ heads,
                              const int*   __restrict__ reservoir,
                              const float* __restrict__ acc,
                              float* __restrict__ out,
                              int N, int R, int n_total) {
    __shared__ float red[128];
    int t = threadIdx.x;
    float v = 0.0f;
    if (t < BATCH * R) {
        int b = t / R;
        int r = t % R;
        int node = reservoir[r];
        float h  = heads[(size_t)b * N + node];
        float d  = h - RES_HEAD_C;
        v = d * d;
    }
    red[t] = v;
    __syncthreads();
    for (int s = 64; s > 0; s >>= 1) {
        if (t < s) red[t] += red[t + s];
        __syncthreads();
    }
    if (t == 0) {
        float bl = red[0] / (float)(BATCH * R);
        float cl = acc[0] / (float)n_total;
        out[0] = cl;        // continuity_loss
        out[1] = bl;        // boundary_loss
        out[2] = cl + bl;   // total_physics_loss (LAMBDA = 1)
    }
}

extern "C" void kernel_launch(void* const* d_in, const int* in_sizes, int n_in,
                              void* d_out, int out_size, void* d_ws, size_t ws_size,
                              hipStream_t stream) {
    const float* heads      = (const float*)d_in[0];  // (B, N)
    const float* demands    = (const float*)d_in[1];  // (B, N)
    const int*   edge_index = (const int*)  d_in[2];  // (2, E)
    const float* edge_attr  = (const float*)d_in[3];  // (E, 2)
    const int*   reservoir  = (const int*)  d_in[4];  // (R,)

    const int BN = in_sizes[0];          // B*N
    const int N  = BN / BATCH;
    const int E  = in_sizes[2] / 2;
    const int R  = in_sizes[4];

    float* ws    = (float*)d_ws;
    float* acc   = ws;                   // ws[0]: continuity sum
    float* net   = ws + 16;              // ws[16 .. 16+BN): net_flows accum
    float* out   = (float*)d_out;        // [cl, bl, total, flows(B*E)]
    float* flows = out + 3;

    // 1) zero accumulators (b128 stores)
    int zn = 16 + BN;
    pinn_zero_ws<<<(zn / 4 + 255) / 256, 256, 0, stream>>>(ws, zn);

    // 2) edge scatter kernel (bandwidth/atomic bound), 2 edges/thread
    int nThreads = (E + 1) / 2;
    pinn_edge_kernel<<<(nThreads + 255) / 256, 256, 0, stream>>>(
        heads, edge_index, edge_attr, flows, net, E, N);

    // 3) continuity loss reduction on the WMMA pipe
    pinn_continuity_wmma<<<256, 256, 0, stream>>>(net, demands, acc, BN);

    // 4) boundary loss + final scalars
    pinn_finalize<<<1, 128, 0, stream>>>(heads, reservoir, acc, out, N, R, BN);
}